// GraphClassifierStats_1949915152976
// MI455X (gfx1250) — compile-verified
//
#include <hip/hip_runtime.h>
#include <hip/hip_bf16.h>

#define N_NODES   100000
#define N_EDGES   1600000
#define N_GRAPHS  512
#define NODE_DIM  64
#define GRAPH_DIM 16
#define HIDDEN    128
#define OUT_DIM   8
#define X_LD      (NODE_DIM + GRAPH_DIM)   // 80

typedef float v2f __attribute__((ext_vector_type(2)));
typedef float v8f __attribute__((ext_vector_type(8)));

// ---------------------------------------------------------------------------
// Degree: deg[dst] += 1 per edge (float atomics; exact below 2^24)
// ---------------------------------------------------------------------------
__global__ void k_deg(const int* __restrict__ dst, float* __restrict__ deg) {
  int e = blockIdx.x * blockDim.x + threadIdx.x;
  if (e >= N_EDGES) return;
  unsafeAtomicAdd(&deg[dst[e]], 1.0f);
}

// dinv[i] = rsqrt(deg[i] + 1), in place
__global__ void k_dinv(float* __restrict__ d) {
  int i = blockIdx.x * blockDim.x + threadIdx.x;
  if (i >= N_NODES) return;
  d[i] = rsqrtf(d[i] + 1.0f);
}

// ---------------------------------------------------------------------------
// WMMA f32 GEMM, LDS-staged:
//   grid: ceil(Mtiles/8) blocks x 256 threads (8 waves).
//   Each wave owns one 16-row M-tile and computes the full N=128 (8 C-tiles,
//   8x v8f accumulators) so the A fragment is loaded once per k-step and
//   reused across 8 WMMAs. B is staged into LDS in 64x128 chunks (32 KB),
//   once per BLOCK instead of once per wave.
// Fragment layouts per ISA 7.12.2 (f32 16x16x4):
//   A: lanes 0-15 K={0,1}, lanes 16-31 K={2,3};  B: same K split, N=lane&15.
//   C/D: VGPR i -> M=i (lanes 0-15) / M=i+8 (lanes 16-31), N=lane&15.
// ---------------------------------------------------------------------------
template <int K, int LDA, bool BIAS, bool RELU>
__global__ void __launch_bounds__(256)
k_wmma_gemm(const float* __restrict__ A, const float* __restrict__ B,
            float* __restrict__ C, const float* __restrict__ bias, int Mtiles) {
  constexpr int KCHUNK = 64;
  __shared__ float Bs[KCHUNK * HIDDEN];            // 32 KB

  const int t    = threadIdx.x;
  const int wave = t >> 5;
  const int lane = t & 31;
  const int half = lane >> 4;                      // 0: K={0,1}, 1: K={2,3}
  const int l15  = lane & 15;

  int mt = blockIdx.x * 8 + wave;
  const bool active = (mt < Mtiles);
  if (!active) mt = Mtiles - 1;                    // clamp: EXEC stays all-1s for WMMA
  const int row0 = mt * 16;
  const float* Arow = A + (size_t)(row0 + l15) * LDA + half * 2;

  v8f acc[8];
#pragma unroll
  for (int nt = 0; nt < 8; ++nt)
    acc[nt] = (v8f){0.f, 0.f, 0.f, 0.f, 0.f, 0.f, 0.f, 0.f};

#pragma unroll
  for (int kc = 0; kc < K; kc += KCHUNK) {
    const int kl = (K - kc < KCHUNK) ? (K - kc) : KCHUNK;   // folds per-chunk
    __syncthreads();                               // protect prior-chunk reads
    for (int i = t * 4; i < kl * HIDDEN; i += 256 * 4)
      *(float4*)&Bs[i] = *(const float4*)&B[kc * HIDDEN + i];   // ldb == HIDDEN
    __syncthreads();

#pragma unroll
    for (int k0 = 0; k0 < kl; k0 += 4) {
      v2f a;
      a.x = Arow[kc + k0 + 0];
      a.y = Arow[kc + k0 + 1];
      const float* brow0 = &Bs[(k0 + half * 2) * HIDDEN + l15];
      const float* brow1 = brow0 + HIDDEN;
#pragma unroll
      for (int nt = 0; nt < 8; ++nt) {
        v2f b;
        b.x = brow0[nt * 16];
        b.y = brow1[nt * 16];
        acc[nt] = __builtin_amdgcn_wmma_f32_16x16x4_f32(
            /*neg_a=*/false, a, /*neg_b=*/false, b,
            /*c_mod=*/(short)0, acc[nt], /*reuse_a=*/false, /*reuse_b=*/false);
      }
    }
  }

  if (active) {                                    // uniform per-wave branch
    float* Crow = C + (size_t)(row0 + half * 8) * HIDDEN;
#pragma unroll
    for (int nt = 0; nt < 8; ++nt) {
      const int col = nt * 16 + l15;
      const float bval = BIAS ? bias[col] : 0.0f;
#pragma unroll
      for (int i = 0; i < 8; ++i) {
        float v = acc[nt][i] + bval;
        if (RELU) v = fmaxf(v, 0.0f);
        Crow[(size_t)i * HIDDEN + col] = v;
      }
    }
  }
}

// ---------------------------------------------------------------------------
// Edge aggregation: agg[dst] += dinv[src]*dinv[dst] * h[src]
// One wave per edge; each lane handles 4 consecutive features (32*4 = 128).
// h (51.2 MB) is L2-resident; atomics resolve at L2.
// ---------------------------------------------------------------------------
__global__ void k_edge_agg(const int* __restrict__ src, const int* __restrict__ dst,
                           const float* __restrict__ dinv, const float* __restrict__ h,
                           float* __restrict__ agg) {
  int gid = blockIdx.x * blockDim.x + threadIdx.x;
  int e = gid >> 5;
  if (e >= N_EDGES) return;
  int lane = threadIdx.x & 31;
  int s = src[e];
  int d = dst[e];
  float w = dinv[s] * dinv[d];
  const float4 v = *(const float4*)(h + (size_t)s * HIDDEN + lane * 4);
  float* o = agg + (size_t)d * HIDDEN + lane * 4;
  unsafeAtomicAdd(o + 0, w * v.x);
  unsafeAtomicAdd(o + 1, w * v.y);
  unsafeAtomicAdd(o + 2, w * v.z);
  unsafeAtomicAdd(o + 3, w * v.w);
}

// out = agg + dinv^2 * h + b   (optional ReLU)
template <bool RELU>
__global__ void k_combine(const float* __restrict__ agg, const float* __restrict__ h,
                          const float* __restrict__ dinv, const float* __restrict__ bias,
                          float* __restrict__ out) {
  size_t gid = (size_t)blockIdx.x * blockDim.x + threadIdx.x;
  if (gid >= (size_t)N_NODES * HIDDEN) return;
  int i = (int)(gid >> 7);
  int f = (int)(gid & 127);
  float di = dinv[i];
  float v = agg[gid] + di * di * h[gid] + bias[f];
  if (RELU) v = fmaxf(v, 0.0f);
  out[gid] = v;
}

// Pool hidden features: poolH[batch[i]][f] += h[i][f]
__global__ void k_pool_h(const float* __restrict__ h, const int* __restrict__ batch,
                         float* __restrict__ poolH) {
  size_t gid = (size_t)blockIdx.x * blockDim.x + threadIdx.x;
  if (gid >= (size_t)N_NODES * HIDDEN) return;
  int i = (int)(gid >> 7);
  int f = (int)(gid & 127);
  unsafeAtomicAdd(&poolH[(size_t)batch[i] * HIDDEN + f], h[gid]);
}

// Pool graph features x[:,64:80] and per-graph node counts
__global__ void k_pool_g(const float* __restrict__ x, const int* __restrict__ batch,
                         float* __restrict__ poolG, float* __restrict__ cnt) {
  int gid = blockIdx.x * blockDim.x + threadIdx.x;
  if (gid >= N_NODES * GRAPH_DIM) return;
  int i = gid >> 4;
  int c = gid & 15;
  int b = batch[i];
  unsafeAtomicAdd(&poolG[b * GRAPH_DIM + c], x[(size_t)i * X_LD + NODE_DIM + c]);
  if (c == 0) unsafeAtomicAdd(&cnt[b], 1.0f);
}

// gcat[g] = [ poolH[g]/max(cnt,1) , poolG[g]/max(cnt,1) ]   (512 x 144)
__global__ void k_gcat(const float* __restrict__ poolH, const float* __restrict__ poolG,
                       const float* __restrict__ cnt, float* __restrict__ gcat) {
  int gid = blockIdx.x * blockDim.x + threadIdx.x;
  if (gid >= N_GRAPHS * (HIDDEN + GRAPH_DIM)) return;
  int g = gid / (HIDDEN + GRAPH_DIM);
  int j = gid - g * (HIDDEN + GRAPH_DIM);
  float inv = 1.0f / fmaxf(cnt[g], 1.0f);
  float v = (j < HIDDEN) ? poolH[g * HIDDEN + j] : poolG[g * GRAPH_DIM + (j - HIDDEN)];
  gcat[gid] = v * inv;
}

// out[g][o] = dot(gh[g], Wm2[:,o]) + bm2[o]   (512 x 8, N too small for WMMA)
__global__ void k_out(const float* __restrict__ gh, const float* __restrict__ Wm2,
                      const float* __restrict__ bm2, float* __restrict__ out) {
  int gid = blockIdx.x * blockDim.x + threadIdx.x;
  if (gid >= N_GRAPHS * OUT_DIM) return;
  int g = gid >> 3;
  int o = gid & 7;
  float acc = bm2[o];
  const float* row = gh + (size_t)g * HIDDEN;
#pragma unroll 8
  for (int k = 0; k < HIDDEN; ++k) acc = fmaf(row[k], Wm2[k * OUT_DIM + o], acc);
  out[gid] = acc;
}

// ---------------------------------------------------------------------------
extern "C" void kernel_launch(void* const* d_in, const int* in_sizes, int n_in,
                              void* d_out, int out_size, void* d_ws, size_t ws_size,
                              hipStream_t stream) {
  (void)in_sizes; (void)n_in; (void)out_size; (void)ws_size;

  const float* x     = (const float*)d_in[0];
  const int*   ei    = (const int*)d_in[1];
  const int*   batch = (const int*)d_in[2];
  const float* W1    = (const float*)d_in[3];
  const float* b1    = (const float*)d_in[4];
  const float* W2    = (const float*)d_in[5];
  const float* b2    = (const float*)d_in[6];
  const float* Wm1   = (const float*)d_in[7];
  const float* bm1   = (const float*)d_in[8];
  const float* Wm2   = (const float*)d_in[9];
  const float* bm2   = (const float*)d_in[10];
  float* out = (float*)d_out;

  const int* srcI = ei;
  const int* dstI = ei + N_EDGES;

  // workspace layout (floats)
  float* ws = (float*)d_ws;
  size_t o = 0;
  float* dinv  = ws + o; o += 100352;                      // deg -> dinv in place
  float* hA    = ws + o; o += (size_t)N_NODES * HIDDEN;    // 12.8M
  float* hB    = ws + o; o += (size_t)N_NODES * HIDDEN;    // 12.8M
  float* poolH = ws + o; o += N_GRAPHS * HIDDEN;
  float* poolG = ws + o; o += N_GRAPHS * GRAPH_DIM;
  float* cnt   = ws + o; o += 512;
  float* gcat  = ws + o; o += N_GRAPHS * (HIDDEN + GRAPH_DIM);
  float* gh    = ws + o; o += N_GRAPHS * HIDDEN;

  const size_t HBYTES = (size_t)N_NODES * HIDDEN * sizeof(float);
  const int MT = N_NODES / 16;                             // 6250 M-tiles

  // zero accumulators (capture-safe async memsets, stream-ordered)
  hipMemsetAsync(dinv, 0, 100352 * sizeof(float), stream);
  hipMemsetAsync(hB, 0, HBYTES, stream);                   // agg1
  hipMemsetAsync(poolH, 0,
                 (N_GRAPHS * HIDDEN + N_GRAPHS * GRAPH_DIM + 512) * sizeof(float),
                 stream);

  // degree / normalization
  k_deg<<<(N_EDGES + 255) / 256, 256, 0, stream>>>(dstI, dinv);
  k_dinv<<<(N_NODES + 255) / 256, 256, 0, stream>>>(dinv);

  // conv1: hA = X[:, :64] @ W1
  k_wmma_gemm<NODE_DIM, X_LD, false, false>
      <<<(MT + 7) / 8, 256, 0, stream>>>(x, W1, hA, nullptr, MT);
  // hB(agg1) += norm * hA[src]
  k_edge_agg<<<(N_EDGES * 32) / 256, 256, 0, stream>>>(srcI, dstI, dinv, hA, hB);
  // hA = relu(hB + dinv^2*hA + b1)
  k_combine<true><<<(N_NODES * HIDDEN) / 256, 256, 0, stream>>>(hB, hA, dinv, b1, hA);

  // conv2: hB = hA @ W2
  k_wmma_gemm<HIDDEN, HIDDEN, false, false>
      <<<(MT + 7) / 8, 256, 0, stream>>>(hA, W2, hB, nullptr, MT);
  hipMemsetAsync(hA, 0, HBYTES, stream);                   // agg2
  k_edge_agg<<<(N_EDGES * 32) / 256, 256, 0, stream>>>(srcI, dstI, dinv, hB, hA);
  // hB = hA + dinv^2*hB + b2   (no relu)
  k_combine<false><<<(N_NODES * HIDDEN) / 256, 256, 0, stream>>>(hA, hB, dinv, b2, hB);

  // pooling
  k_pool_h<<<(N_NODES * HIDDEN) / 256, 256, 0, stream>>>(hB, batch, poolH);
  k_pool_g<<<(N_NODES * GRAPH_DIM) / 256, 256, 0, stream>>>(x, batch, poolG, cnt);
  k_gcat<<<(N_GRAPHS * (HIDDEN + GRAPH_DIM) + 255) / 256, 256, 0, stream>>>(poolH, poolG, cnt, gcat);

  // MLP head: gh = relu(gcat @ Wm1 + bm1)   (32 M-tiles -> 4 blocks)
  k_wmma_gemm<HIDDEN + GRAPH_DIM, HIDDEN + GRAPH_DIM, true, true>
      <<<(N_GRAPHS / 16 + 7) / 8, 256, 0, stream>>>(gcat, Wm1, gh, bm1, N_GRAPHS / 16);
  // out = gh @ Wm2 + bm2
  k_out<<<(N_GRAPHS * OUT_DIM + 255) / 256, 256, 0, stream>>>(gh, Wm2, bm2, out);
}